// loss_separation_28217935134851
// MI455X (gfx1250) — compile-verified
//
#include <hip/hip_runtime.h>
#include <math.h>

// Problem constants (reference: keypoints [32, 2048, 2] f32)
constexpr int BATCH = 32;
constexpr int NPTS  = 2048;
constexpr int NTILE = NPTS / 16;                 // 128 tiles per dimension
constexpr int WAVES_PER_BLOCK = 8;               // 256 threads, wave32
constexpr int TASKS_PER_BATCH = NTILE;           // 64 row-pairs x 2 column parities
constexpr int TOTAL_WAVE_TASKS = BATCH * TASKS_PER_BATCH;      // 4096
constexpr int NBLOCKS = TOTAL_WAVE_TASKS / WAVES_PER_BLOCK;    // 512

typedef __attribute__((ext_vector_type(2))) float v2f;
typedef __attribute__((ext_vector_type(8))) float v8f;

// (0.5 * log2(e))^2 : exp(-0.5*sqrt(d2)) == exp2(-sqrt(d2 * K2))
#define HALF_LOG2E_SQ (0.52034224525140192791f)

// Augmented-K trick: with A row = (x_i, y_i, |x_i|^2, 1) and
// B col = (-2x_j, -2y_j, 1, |x_j|^2), the WMMA D output IS d^2 directly:
//   D[m][n] = -2 x_i x_j - 2 y_i y_j + |x_i|^2 + |x_j|^2
// A layout (16x4 f32): lanes 0-15 -> VGPR0=K0(x), VGPR1=K1(y);
//                      lanes 16-31 -> VGPR0=K2(|x|^2), VGPR1=K3(1).
// B layout (4x16 f32): lanes 0-15 -> VGPR0=K0(-2x), VGPR1=K1(-2y);
//                      lanes 16-31 -> VGPR0=K2(1),   VGPR1=K3(|x|^2).

__device__ __forceinline__ v2f makeRowFrag(const float* __restrict__ base,
                                           int ti, int l16, int half) {
    const int rowIdx = ti * 16 + l16;
    const float ax = base[rowIdx * 2 + 0];
    const float ay = base[rowIdx * 2 + 1];
    const float rn = ax * ax + ay * ay;   // lane L: |row (ti*16 + (L&15))|^2
    v2f a;
    a.x = half ? rn   : ax;
    a.y = half ? 1.0f : ay;
    return a;
}

// Sum of exp(-0.5*dist) over one 16x16 tile (optionally masking the diagonal).
__device__ __forceinline__ float tileSum(v2f amat, float bx, float by,
                                         int half, int l16, bool maskDiag) {
    const float cn = bx * bx + by * by;   // |col (lane&15)|^2 — already per-lane
    v2f bmat;
    bmat.x = half ? 1.0f : (-2.0f * bx);
    bmat.y = half ? cn   : (-2.0f * by);

    v8f c = {};
    c = __builtin_amdgcn_wmma_f32_16x16x4_f32(
            /*neg_a=*/false, amat, /*neg_b=*/false, bmat,
            /*c_mod=*/(short)0, c, /*reuse_a=*/false, /*reuse_b=*/false);

    float s = 0.0f;
#pragma unroll
    for (int r = 0; r < 8; ++r) {
        // c[r] is d^2; |.| (free src modifier) handles tiny negative fp residue,
        // and the negate folds into v_exp_f32's input modifier.
        const float t = c[r] * HALF_LOG2E_SQ;
        float v = __builtin_amdgcn_exp2f(-__builtin_amdgcn_sqrtf(__builtin_fabsf(t)));
        if (maskDiag) {
            const bool diag = ((r + 8 * half) == l16);
            v = diag ? 0.0f : v;
        }
        s += v;
    }
    return s;
}

__global__ __launch_bounds__(256) void pairdist_wmma_kernel(
    const float* __restrict__ kp,   // [BATCH, NPTS, 2]
    float* __restrict__ partial)    // [NBLOCKS]
{
    const int lane = threadIdx.x & 31;
    const int wib  = threadIdx.x >> 5;
    const int wid  = blockIdx.x * WAVES_PER_BLOCK + wib;
    const int b      = wid >> 7;          // 128 tasks per batch
    const int task   = wid & 127;
    const int k      = task >> 1;         // row-pair index 0..63
    const int parity = task & 1;          // column parity split
    const int half = lane >> 4;
    const int l16  = lane & 15;

    const float* __restrict__ base = kp + (size_t)b * NPTS * 2;

    float accOff  = 0.0f;   // tiles with tj > ti (weight 2 by symmetry)
    float accDiag = 0.0f;   // diagonal tiles    (weight 1, diag masked)

    const int rows[2] = { k, NTILE - 1 - k };   // balanced pairing: ~129 tiles/pair

#pragma unroll
    for (int rr = 0; rr < 2; ++rr) {
        const int ti = rows[rr];
        const v2f amat = makeRowFrag(base, ti, l16, half);

        // Diagonal tile: only the parity-0 wave of this (b,k) pair handles it.
        // (parity is wave-uniform -> EXEC is all-1s or the branch is skipped,
        //  so the WMMA below never runs with partial EXEC.)
        if (parity == 0) {
            const int col = ti * 16 + l16;
            accDiag += tileSum(amat, base[col * 2 + 0], base[col * 2 + 1],
                               half, l16, /*maskDiag=*/true);
        }

        // Strict upper-triangle tiles, split by parity; rotated loop so the
        // next tile's load is in flight during this tile's WMMA + trans chain.
        int tj = ti + 1 + parity;
        if (tj < NTILE) {
            int col = tj * 16 + l16;
            float bx = base[col * 2 + 0];
            float by = base[col * 2 + 1];
            while (tj < NTILE) {
                const float cbx = bx, cby = by;
                const int tn = tj + 2;
                if (tn < NTILE) {
                    const int ncol = tn * 16 + l16;
                    bx = base[ncol * 2 + 0];
                    by = base[ncol * 2 + 1];
                }
                accOff += tileSum(amat, cbx, cby, half, l16, /*maskDiag=*/false);
                tj = tn;
            }
        }
    }

    float acc = fmaf(2.0f, accOff, accDiag);

    // ---- deterministic reduction: wave shuffle tree -> LDS -> block partial
#pragma unroll
    for (int off = 16; off > 0; off >>= 1)
        acc += __shfl_xor(acc, off, 32);

    __shared__ float smem[WAVES_PER_BLOCK];
    if (lane == 0) smem[wib] = acc;
    __syncthreads();
    if (threadIdx.x == 0) {
        float s = 0.0f;
#pragma unroll
        for (int w = 0; w < WAVES_PER_BLOCK; ++w) s += smem[w];
        partial[blockIdx.x] = s;
    }
}

// Final deterministic reduction of per-block partials (single block, fixed order).
__global__ __launch_bounds__(256) void reduce_partials_kernel(
    const float* __restrict__ partial, int n, float* __restrict__ out)
{
    __shared__ float smem[256];
    float s = 0.0f;
    for (int i = threadIdx.x; i < n; i += 256) s += partial[i];
    smem[threadIdx.x] = s;
    __syncthreads();
#pragma unroll
    for (int stride = 128; stride > 0; stride >>= 1) {
        if ((int)threadIdx.x < stride) smem[threadIdx.x] += smem[threadIdx.x + stride];
        __syncthreads();
    }
    if (threadIdx.x == 0) out[0] = smem[0];
}

extern "C" void kernel_launch(void* const* d_in, const int* in_sizes, int n_in,
                              void* d_out, int out_size, void* d_ws, size_t ws_size,
                              hipStream_t stream) {
    (void)in_sizes; (void)n_in; (void)out_size; (void)ws_size;
    const float* kp = (const float*)d_in[0];
    float* out      = (float*)d_out;
    float* partial  = (float*)d_ws;   // NBLOCKS floats of scratch

    pairdist_wmma_kernel<<<NBLOCKS, 256, 0, stream>>>(kp, partial);
    reduce_partials_kernel<<<1, 256, 0, stream>>>(partial, NBLOCKS, out);
}